// EMAMultiCodebookQuantizer_27315992003197
// MI455X (gfx1250) — compile-verified
//
#include <hip/hip_runtime.h>
#include <hip/hip_bf16.h>

typedef float v2f __attribute__((ext_vector_type(2)));
typedef float v8f __attribute__((ext_vector_type(8)));

#define B_DIM 2048
#define S_DIM 32
#define K_DIM 1024
#define D_DIM 256

// ---- CDNA5 async global->LDS copy (16B per lane), tracked by ASYNCcnt ----
__device__ __forceinline__ void async_copy_b128(unsigned lds_byte_off,
                                                const float* sbase,
                                                unsigned gbyte_off) {
    asm volatile("global_load_async_to_lds_b128 %0, %1, %2"
                 :: "v"(lds_byte_off), "v"(gbyte_off), "s"(sbase)
                 : "memory");
}

__device__ __forceinline__ void wait_async0() {
#if __has_builtin(__builtin_amdgcn_s_wait_asynccnt)
    __builtin_amdgcn_s_wait_asynccnt(0);
#else
    asm volatile("s_wait_asynccnt 0" ::: "memory");
#endif
}

// -------------------- K0a: e2[s,k] = sum_d emb^2 --------------------
__global__ void k_e2(const float* __restrict__ emb, float* __restrict__ e2) {
    int w = (int)((blockIdx.x * blockDim.x + threadIdx.x) >> 5);  // (s*K + k)
    int lane = threadIdx.x & 31;
    if (w >= S_DIM * K_DIM) return;
    const float* er = emb + (size_t)w * D_DIM;
    float a = 0.f;
#pragma unroll
    for (int j = 0; j < 8; ++j) { float x = er[lane + j * 32]; a += x * x; }
#pragma unroll
    for (int m = 16; m >= 1; m >>= 1) a += __shfl_xor(a, m, 32);
    if (lane == 0) e2[w] = a;
}

// -------------------- K0b: out[i] = 0.99 * in[i] --------------------
__global__ void k_scale(const float* __restrict__ in, float* __restrict__ out, int n) {
    const float DEC = (float)0.99;
    int i = blockIdx.x * blockDim.x + threadIdx.x;
    int stride = gridDim.x * blockDim.x;
    for (; i < n; i += stride) out[i] = DEC * in[i];
}

// -------------------- K1: fused distance GEMM (WMMA f32) + argmin --------------------
__launch_bounds__(256)
__global__ void k_dist_argmin(const float* __restrict__ z,
                              const float* __restrict__ emb,
                              const float* __restrict__ e2,
                              int* __restrict__ idx_out,
                              float* __restrict__ mind_out) {
    const int s = blockIdx.y;
    const int btile = blockIdx.x;          // 16 tiles of 128 rows
    const int t = threadIdx.x;             // 256 threads = 8 waves
    const int lane = t & 31;
    const int wave = t >> 5;
    const int l15 = lane & 15;
    const int lhi = lane >> 4;             // 0 or 1

    // double-buffered staged embedding chunk: 2 x (32 codes x 256 d), padded stride 260
    __shared__ float lds_e[2][32 * 260];
    __shared__ float z2s[128];

    const int rbase = btile * 128;
    const int wrow = rbase + wave * 16;

    // z^2 per row (2 threads per row, halves of D)
    {
        int r = t >> 1;
        int half = t & 1;
        const float* zr = z + ((size_t)(rbase + r) * S_DIM + s) * D_DIM + half * 128;
        float a = 0.f;
#pragma unroll 8
        for (int j = 0; j < 128; ++j) a += zr[j] * zr[j];
        a += __shfl_xor(a, 1, 32);
        if (half == 0) z2s[r] = a;
    }

    // Preload A fragments for the whole contraction (D=256 -> 64 k4-steps).
    // 32-bit A 16x4 layout: lane(M=l15), VGPR0=K(d0+2*lhi), VGPR1=K+1 -> float2 load.
    v2f afrag[64];
    {
        const float* za = z + ((size_t)(wrow + l15) * S_DIM + s) * D_DIM + lhi * 2;
#pragma unroll
        for (int j = 0; j < 64; ++j) afrag[j] = *(const v2f*)(za + j * 4);
    }

    float minv[8];
    int mini[8];
#pragma unroll
    for (int r = 0; r < 8; ++r) { minv[r] = 3.4e38f; mini[r] = 0; }

    const float* es = emb + (size_t)s * K_DIM * D_DIM;
    const float* e2s = e2 + (size_t)s * K_DIM;

    // per-thread staging geometry: 8 rounds of 16B, 32 codes x 1KB rows
    const unsigned lds_base = (unsigned)(uintptr_t)&lds_e[0][0];

    // prefetch chunk 0 into buffer 0 (async, no VGPR bounce)
    {
        const float* src = es;
#pragma unroll
        for (int it = 0; it < 8; ++it) {
            int i4 = t + it * 256;
            int r = i4 >> 6;
            int c = (i4 & 63) * 4;
            async_copy_b128(lds_base + (unsigned)((r * 260 + c) * 4),
                            src, (unsigned)((r * D_DIM + c) * 4));
        }
    }
    wait_async0();
    __syncthreads();

    for (int nb = 0; nb < K_DIM / 32; ++nb) {
        const int cur = nb & 1;

        // async prefetch of next chunk into the other buffer (overlaps WMMAs)
        if (nb + 1 < K_DIM / 32) {
            const float* src = es + (size_t)((nb + 1) * 32) * D_DIM;
            const unsigned bufoff = lds_base + (unsigned)((cur ^ 1) * 32 * 260 * 4);
#pragma unroll
            for (int it = 0; it < 8; ++it) {
                int i4 = t + it * 256;
                int r = i4 >> 6;
                int c = (i4 & 63) * 4;
                async_copy_b128(bufoff + (unsigned)((r * 260 + c) * 4),
                                src, (unsigned)((r * D_DIM + c) * 4));
            }
        }

        v8f acc0 = (v8f)0.0f;
        v8f acc1 = (v8f)0.0f;
        // B 4x16 layout: lane(N=l15), VGPR0=K(d0+2*lhi), VGPR1=K+1 -> float2 from LDS
        const float* b0 = &lds_e[cur][(0 * 16 + l15) * 260 + lhi * 2];
        const float* b1 = &lds_e[cur][(1 * 16 + l15) * 260 + lhi * 2];
#pragma unroll
        for (int j = 0; j < 64; ++j) {
            v2f bf0 = *(const v2f*)(b0 + j * 4);
            v2f bf1 = *(const v2f*)(b1 + j * 4);
            acc0 = __builtin_amdgcn_wmma_f32_16x16x4_f32(false, afrag[j], false, bf0,
                                                         (short)0, acc0, false, false);
            acc1 = __builtin_amdgcn_wmma_f32_16x16x4_f32(false, afrag[j], false, bf1,
                                                         (short)0, acc1, false, false);
        }

        // next buffer must be complete (this wave) and all waves done reading cur
        wait_async0();
        __syncthreads();

        // epilogue: score = e2 - 2*ze  (z2 is row-constant; added at the end)
        int col0 = nb * 32 + l15;
        int col1 = col0 + 16;
        float e20 = e2s[col0];
        float e21 = e2s[col1];
#pragma unroll
        for (int r = 0; r < 8; ++r) {
            float s0 = e20 - 2.0f * acc0[r];
            float s1 = e21 - 2.0f * acc1[r];
            if (s0 < minv[r]) { minv[r] = s0; mini[r] = col0; }
            if (s1 < minv[r]) { minv[r] = s1; mini[r] = col1; }
        }
    }

    // reduce (min,argmin) across the 16 lanes of each half-wave
#pragma unroll
    for (int r = 0; r < 8; ++r) {
        float v = minv[r];
        int i = mini[r];
#pragma unroll
        for (int m = 1; m <= 8; m <<= 1) {
            float ov = __shfl_xor(v, m, 32);
            int oi = __shfl_xor(i, m, 32);
            if (ov < v || (ov == v && oi < i)) { v = ov; i = oi; }
        }
        if (l15 == 0) {
            int row = lhi * 8 + r;                 // C/D layout: M = r + 8*(lane>>4)
            int b = wrow + row;
            size_t o = (size_t)b * S_DIM + s;
            idx_out[o] = i;
            mind_out[o] = z2s[wave * 16 + row] + v; // = ||z - e||^2
        }
    }
}

// -------------------- K2: gather z_q + EMA scatter --------------------
__global__ void k_scatter(const float* __restrict__ z,
                          const float* __restrict__ emb,
                          const int* __restrict__ idx,
                          float* __restrict__ zq_out,
                          float* __restrict__ idxf_out,
                          float* __restrict__ ecs_out,
                          float* __restrict__ emaw_out) {
    int w = (int)((blockIdx.x * blockDim.x + threadIdx.x) >> 5);  // (b*S + s)
    int lane = threadIdx.x & 31;
    if (w >= B_DIM * S_DIM) return;
    int s = w & (S_DIM - 1);
    int k = idx[w];
    const float ONE_M = (float)(1.0 - 0.99);
    const float* zr = z + (size_t)w * D_DIM;
    const float* er = emb + ((size_t)s * K_DIM + k) * D_DIM;
    float* ew = emaw_out + ((size_t)s * K_DIM + k) * D_DIM;
    float* zq = zq_out + (size_t)w * D_DIM;
#pragma unroll
    for (int j = 0; j < 8; ++j) {
        int d = lane + j * 32;
        float zv = zr[d];
        float ev = er[d];
        zq[d] = zv + (ev - zv);                 // straight-through value
        atomicAdd(&ew[d], ONE_M * zv);          // dw accumulation
    }
    if (lane == 0) {
        atomicAdd(&ecs_out[s * K_DIM + k], ONE_M);
        idxf_out[w] = (float)k;
    }
}

// -------------------- K3a: n[s] (blocks 0..31) and loss (block 32) --------------------
__global__ void k_reduce(const float* __restrict__ ecs_out,
                         const float* __restrict__ mind,
                         float* __restrict__ n_ws,
                         float* __restrict__ loss_out) {
    __shared__ float sm[256];
    int t = threadIdx.x;
    float a = 0.f;
    if ((int)blockIdx.x < S_DIM) {
        int s = blockIdx.x;
        for (int k = t; k < K_DIM; k += 256) a += ecs_out[s * K_DIM + k];
    } else {
        for (int i = t; i < B_DIM * S_DIM; i += 256) a += mind[i];
    }
    sm[t] = a;
    __syncthreads();
    for (int off = 128; off >= 1; off >>= 1) {
        if (t < off) sm[t] += sm[t + off];
        __syncthreads();
    }
    if (t == 0) {
        if ((int)blockIdx.x < S_DIM) n_ws[blockIdx.x] = sm[0];
        else loss_out[0] = sm[0] / (float)((size_t)B_DIM * S_DIM * D_DIM);
    }
}

// -------------------- K3b: new_embedding = new_ema_w / cs --------------------
__global__ void k_newemb(const float* __restrict__ emaw_out,
                         const float* __restrict__ ecs_out,
                         const float* __restrict__ n_ws,
                         float* __restrict__ emb_out) {
    const float EPSF = 1e-5f;
    size_t i = (size_t)blockIdx.x * blockDim.x + threadIdx.x;
    size_t total = (size_t)S_DIM * K_DIM * D_DIM;
    if (i >= total) return;
    size_t sk = i >> 8;                 // / D
    float ecs = ecs_out[sk];
    float nn = n_ws[sk >> 10];          // / K
    float cs = (ecs + EPSF) / (nn + (float)K_DIM * EPSF) * nn;
    emb_out[i] = emaw_out[i] / cs;
}

extern "C" void kernel_launch(void* const* d_in, const int* in_sizes, int n_in,
                              void* d_out, int out_size, void* d_ws, size_t ws_size,
                              hipStream_t stream) {
    const float* z    = (const float*)d_in[0];  // [B,S,D]
    const float* emb  = (const float*)d_in[1];  // [S,K,D]
    const float* ecs  = (const float*)d_in[2];  // [S,K]
    const float* emaw = (const float*)d_in[3];  // [S,K,D]
    float* out = (float*)d_out;

    const size_t N_ZQ = (size_t)B_DIM * S_DIM * D_DIM;   // 16777216
    const size_t N_BS = (size_t)B_DIM * S_DIM;           // 65536
    const size_t N_SKD = (size_t)S_DIM * K_DIM * D_DIM;  // 8388608
    const size_t N_SK = (size_t)S_DIM * K_DIM;           // 32768

    float* out_zq   = out;
    float* out_loss = out_zq + N_ZQ;
    float* out_idx  = out_loss + 1;
    float* out_emb  = out_idx + N_BS;
    float* out_ecs  = out_emb + N_SKD;
    float* out_emaw = out_ecs + N_SK;

    float* ws      = (float*)d_ws;
    float* e2_ws   = ws;                         // 32768 f32
    int*   idx_ws  = (int*)(ws + N_SK);          // 65536 i32
    float* mind_ws = ws + N_SK + N_BS;           // 65536 f32
    float* n_ws    = mind_ws + N_BS;             // 32 f32

    // K0: precompute e2, seed EMA output regions with decay*old
    k_e2<<<(int)(N_SK * 32 / 256), 256, 0, stream>>>(emb, e2_ws);
    k_scale<<<(int)((N_SK + 255) / 256), 256, 0, stream>>>(ecs, out_ecs, (int)N_SK);
    k_scale<<<8192, 256, 0, stream>>>(emaw, out_emaw, (int)N_SKD);

    // K1: WMMA distance GEMM + fused argmin (async double-buffered staging)
    dim3 g1(B_DIM / 128, S_DIM);
    k_dist_argmin<<<g1, 256, 0, stream>>>(z, emb, e2_ws, idx_ws, mind_ws);

    // K2: gather z_q + EMA scatter (atomics into seeded output regions)
    k_scatter<<<(int)(N_BS * 32 / 256), 256, 0, stream>>>(z, emb, idx_ws, out_zq,
                                                          out_idx, out_ecs, out_emaw);

    // K3: per-codebook sums + loss, then normalized embedding
    k_reduce<<<S_DIM + 1, 256, 0, stream>>>(out_ecs, mind_ws, n_ws, out_loss);
    k_newemb<<<(int)((N_SKD + 255) / 256), 256, 0, stream>>>(out_emaw, out_ecs, n_ws, out_emb);
}